// CapsuleLayer_for_final_model_57707180589543
// MI455X (gfx1250) — compile-verified
//
#include <hip/hip_runtime.h>
#include <math.h>

// CDNA5 / gfx1250 capsule routing layer, fused single kernel.
// fp32 WMMA (V_WMMA_F32_16X16X4_F32) computing UhatT = W x u^T so that
// the routing dot-product over i is IN-LANE (register axis), leaving only
// one cross-half shuffle per (n,k). u_hat is recomputed per routing
// iteration out of L2 (inputs 134MB < 192MB L2) instead of spilling
// 268MB of u_hat to HBM.

typedef __attribute__((ext_vector_type(2))) float v2f;
typedef __attribute__((ext_vector_type(8))) float v8f;

#define CAPS_N_IN   512
#define CAPS_N_OUT  2
#define CAPS_D      16
#define CAPS_ROUT   3
#define CAPS_EPS    1e-7f
#define CAPS_BT     16   // batch tile per workgroup (WMMA N)
#define CAPS_WAVES  8    // 256 threads
#define CAPS_NPW    (CAPS_N_IN / CAPS_WAVES)   // 64 n per wave

__global__ __launch_bounds__(256)
void caps_routing_kernel(const float* __restrict__ inputs,  // [B,512,16]
                         const float* __restrict__ W,       // [512,2,16,16] (i,j)
                         float* __restrict__ out)           // [B,2,16]
{
    // LDS scratch (~20.2 KB total -> high occupancy)
    __shared__ float s_buf[CAPS_WAVES][CAPS_N_OUT][CAPS_BT][CAPS_D]; // [w][k][b][i] 16 KB
    __shared__ float s_fin[CAPS_N_OUT][CAPS_BT][CAPS_D];             // [k][b][i] 2 KB
    __shared__ float vacc [CAPS_N_OUT][CAPS_BT][CAPS_D];             // [k][b][i] 2 KB
    __shared__ float fct  [CAPS_N_OUT][CAPS_BT];                     // squash factors

    const int t    = threadIdx.x;
    const int lane = t & 31;
    const int wv   = t >> 5;        // wave id 0..7
    const int half = lane >> 4;     // lane half (0/1)
    const int l16  = lane & 15;
    const int b0   = blockIdx.x * CAPS_BT;

    // zero Vacc (512 floats, 2 per thread)
    ((float*)vacc)[t]       = 0.0f;
    ((float*)vacc)[t + 256] = 0.0f;
    __syncthreads();

    const int n_lo = wv * CAPS_NPW;

    // B-fragment (u^T, K=j, N=b): lane holds batch column b = l16,
    // K-pair by half: b.x/b.y cover j = kk*4 + 2*half + {0,1} (contiguous).
    const float* Urow = inputs + (size_t)(b0 + l16) * CAPS_N_IN * CAPS_D + half * 2;

    for (int it = 0; it < CAPS_ROUT; ++it) {
        // Vacc fragments in C/D layout of UhatT: element r -> i = r+8*half, lane -> b = l16
        float vk0[8], vk1[8];
#pragma unroll
        for (int r = 0; r < 8; ++r) {
            vk0[r] = vacc[0][l16][r + 8 * half];
            vk1[r] = vacc[1][l16][r + 8 * half];
        }

        v8f s0 = {}; // partial s for k=0, layout [i=r+8h][b=l16]
        v8f s1 = {}; // partial s for k=1

        for (int n = n_lo; n < n_lo + CAPS_NPW; ++n) {
            // B fragments (u^T) for this n, shared by k=0 and k=1
            v2f bu[4];
#pragma unroll
            for (int kk = 0; kk < 4; ++kk)
                bu[kk] = *(const v2f*)(Urow + n * CAPS_D + kk * 4);

            // A fragments: W[n][k] with M = i = l16, K-pair j by half (contiguous float2)
            const float* Wa0 = W + (((size_t)n * CAPS_N_OUT + 0) * CAPS_D + l16) * CAPS_D + half * 2;
            const float* Wa1 = W + (((size_t)n * CAPS_N_OUT + 1) * CAPS_D + l16) * CAPS_D + half * 2;

            v8f acc0 = {};   // UhatT[i, b] for k=0
#pragma unroll
            for (int kk = 0; kk < 4; ++kk) {
                v2f aw = *(const v2f*)(Wa0 + kk * 4);
                acc0 = __builtin_amdgcn_wmma_f32_16x16x4_f32(
                    false, aw, false, bu[kk], (short)0, acc0, false, false);
            }
            v8f acc1 = {};   // UhatT[i, b] for k=1
#pragma unroll
            for (int kk = 0; kk < 4; ++kk) {
                v2f aw = *(const v2f*)(Wa1 + kk * 4);
                acc1 = __builtin_amdgcn_wmma_f32_16x16x4_f32(
                    false, aw, false, bu[kk], (short)0, acc1, false, false);
            }

            // logits: in-lane partial dot over i = r+8*half, then one cross-half add
            float p0 = 0.0f, p1 = 0.0f;
#pragma unroll
            for (int r = 0; r < 8; ++r) {
                p0 = fmaf(acc0[r], vk0[r], p0);
                p1 = fmaf(acc1[r], vk1[r], p1);
            }
            p0 += __shfl_xor(p0, 16, 32);
            p1 += __shfl_xor(p1, 16, 32);

            // 2-way softmax over k (per-lane scalar: one exp per n)
            float c0 = 1.0f / (1.0f + __expf(p1 - p0));
            float c1 = 1.0f - c0;
#pragma unroll
            for (int r = 0; r < 8; ++r) {
                s0[r] += c0 * acc0[r];
                s1[r] += c1 * acc1[r];
            }
        }

        // stash per-wave partial s: [w][k][b=l16][i=r+8h]
#pragma unroll
        for (int r = 0; r < 8; ++r) {
            s_buf[wv][0][l16][r + 8 * half] = s0[r];
            s_buf[wv][1][l16][r + 8 * half] = s1[r];
        }
        __syncthreads();

        // reduce 8 wave partials: 512 (k,b,i) combos, 2 per thread
#pragma unroll
        for (int q = 0; q < 2; ++q) {
            int c = t + q * 256;
            int k = c >> 8, b = (c >> 4) & 15, i = c & 15;
            float acc = 0.0f;
#pragma unroll
            for (int w = 0; w < CAPS_WAVES; ++w) acc += s_buf[w][k][b][i];
            s_fin[k][b][i] = acc;
        }
        __syncthreads();

        // squash factor per (k,b): f = |s|^2 / ((1+|s|^2) * (|s| + eps))
        if (t < CAPS_N_OUT * CAPS_BT) {
            int k = t >> 4, b = t & 15;
            float sq = 0.0f;
#pragma unroll
            for (int i = 0; i < CAPS_D; ++i) {
                float x = s_fin[k][b][i];
                sq += x * x;
            }
            float nrm = sqrtf(sq);
            fct[k][b] = sq / ((1.0f + sq) * (nrm + CAPS_EPS));
        }
        __syncthreads();

        // v = f * s ; Vacc += v ; last iteration writes output
#pragma unroll
        for (int q = 0; q < 2; ++q) {
            int c = t + q * 256;
            int k = c >> 8, b = (c >> 4) & 15, i = c & 15;
            float nv = fct[k][b] * s_fin[k][b][i];
            vacc[k][b][i] += nv;
            if (it == CAPS_ROUT - 1)
                out[(size_t)(b0 + b) * (CAPS_N_OUT * CAPS_D) + k * CAPS_D + i] = nv;
        }
        __syncthreads();
    }
}

extern "C" void kernel_launch(void* const* d_in, const int* in_sizes, int n_in,
                              void* d_out, int out_size, void* d_ws, size_t ws_size,
                              hipStream_t stream) {
    const float* inputs = (const float*)d_in[0]; // [B,8,8,8,16] = [B,512,16]
    const float* W      = (const float*)d_in[1]; // [1,512,2,16,16]
    float* out          = (float*)d_out;         // [B,1,2,16]

    int B = in_sizes[0] / (CAPS_N_IN * CAPS_D);  // 4096
    int blocks = B / CAPS_BT;                    // 256
    caps_routing_kernel<<<blocks, 256, 0, stream>>>(inputs, W, out);
}